// CloakBlock_22265110462469
// MI455X (gfx1250) — compile-verified
//
#include <hip/hip_runtime.h>
#include <hip/hip_bf16.h>
#include <math.h>

// ---------------------------------------------------------------------------
// CloakBlock: cosine-similarity gated row replacement.
//   inputs : original [1,512,2048,128] f32, styled [1,512,2048,128] f32
//   per row r (C*H = 1,048,576 rows of W=128):
//     score = dot(o,s) / (max(|o|,eps)*max(|s|,eps))
//     mask  = (0.17 < score < 0.29) && C!=0 && H!=0
//     out[r] = mask ? styled[r] : original[r]
// Memory bound: 1.5 GB @ 23.3 TB/s ~= 64 us. WMMA f32 16x16x4 performs the
// three row reductions (diagonals of O*S^T, O*O^T, S*S^T); async b128 loads
// stage tiles into LDS; NT b128 stores stream the output.
// ---------------------------------------------------------------------------

typedef float v2f __attribute__((ext_vector_type(2)));
typedef float v4f __attribute__((ext_vector_type(4)));
typedef float v8f __attribute__((ext_vector_type(8)));
typedef int   v4i_t __attribute__((vector_size(16)));   // matches builtin pointee

#define ROWS_PER_WAVE   16
#define WAVES_PER_BLOCK 4
#define ROWS_PER_BLOCK  (ROWS_PER_WAVE * WAVES_PER_BLOCK)   // 64
#define WIDTH           128                                  // vector length
#define HDIM            2048
#define CDIM            512
#define TOTAL_ROWS      (CDIM * HDIM)                        // 1,048,576

#if __has_builtin(__builtin_amdgcn_global_load_async_to_lds_b128)
#define HAVE_ASYNC_LDS 1
#else
#define HAVE_ASYNC_LDS 0
#endif

// Builtin signature (from hipcc diagnostic): arg0 = global src as v4i*, AS1;
// arg1 = LDS dst as v4i*, AS3; then imm offset, imm cpol.
#define AS1V4(p) ((__attribute__((address_space(1))) v4i_t*)(void*)(p))
#define AS3V4(p) ((__attribute__((address_space(3))) v4i_t*)(p))

__device__ __forceinline__ float readlane_f(float v, int lane) {
    return __int_as_float(__builtin_amdgcn_readlane(__float_as_int(v), lane));
}

__global__ __launch_bounds__(WAVES_PER_BLOCK * 32)
void cloak_wmma_kernel(const float* __restrict__ o,
                       const float* __restrict__ s,
                       float* __restrict__ out) {
    __shared__ float oT[ROWS_PER_BLOCK * WIDTH];   // 32 KB
    __shared__ float sT[ROWS_PER_BLOCK * WIDTH];   // 32 KB

    const int lane = threadIdx.x & 31;
    const int wave = threadIdx.x >> 5;

    const int    rowBase = blockIdx.x * ROWS_PER_BLOCK + wave * ROWS_PER_WAVE;
    const size_t goff    = (size_t)rowBase * WIDTH;          // float offset

    float* oSlice = &oT[wave * ROWS_PER_WAVE * WIDTH];
    float* sSlice = &sT[wave * ROWS_PER_WAVE * WIDTH];

    // ---- stage 16 rows (8 KB each tensor) into this wave's LDS slice ------
#if HAVE_ASYNC_LDS
    #pragma unroll
    for (int j = 0; j < 16; ++j) {
        const int off = j * 128 + lane * 4;                  // floats
        __builtin_amdgcn_global_load_async_to_lds_b128(
            AS1V4(o + goff + off), AS3V4(oSlice + off), 0, 0);
        __builtin_amdgcn_global_load_async_to_lds_b128(
            AS1V4(s + goff + off), AS3V4(sSlice + off), 0, 0);
    }
#if __has_builtin(__builtin_amdgcn_s_wait_asynccnt)
    __builtin_amdgcn_s_wait_asynccnt(0);
#else
    asm volatile("s_wait_asynccnt 0" ::: "memory");
#endif
#else
    #pragma unroll
    for (int j = 0; j < 16; ++j) {
        const int off = j * 128 + lane * 4;
        *(v4f*)(oSlice + off) = *(const v4f*)(o + goff + off);
        *(v4f*)(sSlice + off) = *(const v4f*)(s + goff + off);
    }
#endif

    // ---- WMMA reductions: diag(O S^T)=dot, diag(O O^T)=|o|^2, diag(S S^T)=|s|^2
    // A(16x4) lane map: (M = lane%16, K = 2*(lane/16)+vgpr)
    // B(4x16) lane map: (N = lane%16, K = 2*(lane/16)+vgpr)  -> identical loads.
    const int r  = lane & 15;
    const int hi = lane >> 4;
    const float* oa = oSlice + r * WIDTH + 2 * hi;
    const float* sb = sSlice + r * WIDTH + 2 * hi;

    v8f c_os = {0.f,0.f,0.f,0.f,0.f,0.f,0.f,0.f};
    v8f c_oo = {0.f,0.f,0.f,0.f,0.f,0.f,0.f,0.f};
    v8f c_ss = {0.f,0.f,0.f,0.f,0.f,0.f,0.f,0.f};

    #pragma unroll
    for (int k0 = 0; k0 < WIDTH; k0 += 4) {
        v2f a = *(const v2f*)(oa + k0);   // ds_load_b64
        v2f b = *(const v2f*)(sb + k0);   // ds_load_b64
        c_os = __builtin_amdgcn_wmma_f32_16x16x4_f32(
            false, a, false, b, (short)0, c_os, false, false);
        c_oo = __builtin_amdgcn_wmma_f32_16x16x4_f32(
            false, a, false, a, (short)0, c_oo, false, false);
        c_ss = __builtin_amdgcn_wmma_f32_16x16x4_f32(
            false, b, false, b, (short)0, c_ss, false, false);
    }

    // ---- extract diagonals, build per-row replacement mask (uniform) ------
    // diag (rr,rr): VGPR rr&7; lane rr (rr<8) or rr+16 (rr>=8).
    unsigned m16 = 0;
    #pragma unroll
    for (int rr = 0; rr < 16; ++rr) {
        const int ln = (rr & 8) ? (rr + 16) : rr;
        const float d  = readlane_f(c_os[rr & 7], ln);
        const float oo = readlane_f(c_oo[rr & 7], ln);
        const float ss = readlane_f(c_ss[rr & 7], ln);
        const float na = fmaxf(sqrtf(oo), 1e-8f);
        const float nb = fmaxf(sqrtf(ss), 1e-8f);
        const float sc = d / (na * nb);
        const int   R  = rowBase + rr;
        const bool edge = ((R >> 11) == 0) || ((R & (HDIM - 1)) == 0); // C==0 || H==0
        if ((sc > 0.17f) && (sc < 0.29f) && !edge) m16 |= (1u << rr);
    }

    // ---- stream out: selected row from LDS, non-temporal b128 stores ------
    #pragma unroll
    for (int rr = 0; rr < 16; ++rr) {
        const float* src = ((m16 >> rr) & 1u) ? (sSlice + rr * WIDTH)
                                              : (oSlice + rr * WIDTH);
        v4f v = *(const v4f*)(src + (lane << 2));            // ds_load_b128
        __builtin_nontemporal_store(
            v, (v4f*)(out + goff + rr * WIDTH + (lane << 2)));
    }
}

extern "C" void kernel_launch(void* const* d_in, const int* in_sizes, int n_in,
                              void* d_out, int out_size, void* d_ws, size_t ws_size,
                              hipStream_t stream) {
    (void)in_sizes; (void)n_in; (void)out_size; (void)d_ws; (void)ws_size;
    const float* original = (const float*)d_in[0];
    const float* styled   = (const float*)d_in[1];
    float*       out      = (float*)d_out;

    const int blocks = TOTAL_ROWS / ROWS_PER_BLOCK;          // 16384
    cloak_wmma_kernel<<<blocks, WAVES_PER_BLOCK * 32, 0, stream>>>(
        original, styled, out);
}